// SVD_45131516346378
// MI455X (gfx1250) — compile-verified
//
#include <hip/hip_runtime.h>

typedef __attribute__((ext_vector_type(2))) float v2f;
typedef __attribute__((ext_vector_type(8))) float v8f;

#define THREADS 256

__global__ __launch_bounds__(THREADS) void kabsch2d_kernel(
    const float* __restrict__ src, const float* __restrict__ tgt,
    const float* __restrict__ wts, const int* __restrict__ convert_flag,
    float* __restrict__ out, int B, int N)
{
    const int b   = blockIdx.x;
    const int tid = threadIdx.x;
    const float res  = 0.2592f;
    const float cmin = (640.0f * 0.5f - 0.5f) * 0.2592f;   // CART_PIXEL_WIDTH even
    const bool conv = (convert_flag[0] != 0);

    const float4* __restrict__ src4 = (const float4*)(src + (size_t)b * N * 2);
    const float4* __restrict__ tgt4 = (const float4*)(tgt + (size_t)b * N * 2);
    const float4* __restrict__ w4   = (const float4*)(wts + (size_t)b * N);

    // 9 raw weighted sums: w, w*s, w*t, w*(t outer s)
    float aW = 0.f, aSx = 0.f, aSy = 0.f, aTx = 0.f, aTy = 0.f;
    float aXX = 0.f, aXY = 0.f, aYX = 0.f, aYY = 0.f;

    const int nq = N >> 2;  // 4 points per quad
    for (int q = tid; q < nq; q += THREADS) {
        float4 s0 = src4[2 * q], s1 = src4[2 * q + 1];
        float4 t0 = tgt4[2 * q], t1 = tgt4[2 * q + 1];
        float4 w  = w4[q];
        if (q + THREADS < nq) {   // streaming prefetch -> global_prefetch_b8
            __builtin_prefetch(&src4[2 * (q + THREADS)], 0, 0);
            __builtin_prefetch(&tgt4[2 * (q + THREADS)], 0, 0);
            __builtin_prefetch(&w4[q + THREADS], 0, 0);
        }
        float sx[4] = {s0.x, s0.z, s1.x, s1.z};
        float sy[4] = {s0.y, s0.w, s1.y, s1.w};
        float tx[4] = {t0.x, t0.z, t1.x, t1.z};
        float ty[4] = {t0.y, t0.w, t1.y, t1.w};
        float ww[4] = {w.x, w.y, w.z, w.w};
        #pragma unroll
        for (int k = 0; k < 4; ++k) {
            float msx, msy, mtx, mty;
            if (conv) {
                msx = cmin - res * sy[k];  msy = res * sx[k] - cmin;
                mtx = cmin - res * ty[k];  mty = res * tx[k] - cmin;
            } else {
                msx = sx[k]; msy = sy[k]; mtx = tx[k]; mty = ty[k];
            }
            float wk = ww[k];
            aW += wk;
            aSx = fmaf(wk, msx, aSx);  aSy = fmaf(wk, msy, aSy);
            float wtx = wk * mtx, wty = wk * mty;
            aTx += wtx;  aTy += wty;
            aXX = fmaf(wtx, msx, aXX); aXY = fmaf(wtx, msy, aXY);
            aYX = fmaf(wty, msx, aYX); aYY = fmaf(wty, msy, aYY);
        }
    }

    // ---- wave32 butterfly reduction of the 9 accumulators ----
    float acc[9] = {aW, aSx, aSy, aTx, aTy, aXX, aXY, aYX, aYY};
    #pragma unroll
    for (int j = 0; j < 9; ++j) {
        #pragma unroll
        for (int off = 16; off > 0; off >>= 1)
            acc[j] += __shfl_xor(acc[j], off, 32);
    }

    __shared__ float part[8][9];   // 8 waves x 9 partials
    __shared__ float sums[9];
    const int wave = tid >> 5, lane = tid & 31;
    if (lane == 0) {
        #pragma unroll
        for (int j = 0; j < 9; ++j) part[wave][j] = acc[j];
    }
    __syncthreads();

    // ---- cross-wave reduction with V_WMMA_F32_16X16X4_F32 (B = ones) ----
    if (tid < 32) {
#if __has_builtin(__builtin_amdgcn_wmma_f32_16x16x4_f32)
        // A (16x4 f32): lanes 0-15 row M=lane, VGPR0=K0,VGPR1=K1;
        //               lanes 16-31 same rows, VGPR0=K2,VGPR1=K3.
        int row = tid & 15, hi = tid >> 4;
        v2f A0, A1, Bv;
        A0.x = (row < 9) ? part[2 * hi + 0][row] : 0.0f;  // waves 0..3
        A0.y = (row < 9) ? part[2 * hi + 1][row] : 0.0f;
        A1.x = (row < 9) ? part[2 * hi + 4][row] : 0.0f;  // waves 4..7
        A1.y = (row < 9) ? part[2 * hi + 5][row] : 0.0f;
        Bv.x = 1.0f; Bv.y = 1.0f;                         // all-ones B: layout-agnostic
        v8f C = {};
        C = __builtin_amdgcn_wmma_f32_16x16x4_f32(false, A0, false, Bv,
                                                  (short)0, C, false, false);
        C = __builtin_amdgcn_wmma_f32_16x16x4_f32(false, A1, false, Bv,
                                                  (short)0, C, false, false);
        // D[m][0]: lane 0 holds m=0..7 in C[0..7]; lane 16 holds m=8 in C[0]
        if (tid == 0) {
            #pragma unroll
            for (int j = 0; j < 8; ++j) sums[j] = C[j];
        }
        if (tid == 16) sums[8] = C[0];
#else
        if (tid == 0) {
            for (int j = 0; j < 9; ++j) {
                float s = 0.f;
                for (int wv = 0; wv < 8; ++wv) s += part[wv][j];
                sums[j] = s;
            }
        }
#endif
    }
    __syncthreads();

    // ---- per-batch epilogue: closed-form 2x2 polar factor == 3x3 SVD path ----
    if (tid == 0) {
        float Sw  = sums[0];
        float Ssx = sums[1], Ssy = sums[2];
        float Stx = sums[3], Sty = sums[4];
        float Sxx = sums[5], Sxy = sums[6], Syx = sums[7], Syy = sums[8];

        float wsum = Sw + 0.0001f;
        float inv  = 1.0f / wsum;
        float scx = Ssx * inv, scy = Ssy * inv;
        float tcx = Stx * inv, tcy = Sty * inv;

        // W[d][e] = (S_ts - tc_d*S_s_e - sc_e*S_t_d + tc_d*sc_e*Sw) / wsum
        float a  = (Sxx - tcx * Ssx - scx * Stx + tcx * scx * Sw) * inv;
        float bb = (Sxy - tcx * Ssy - scy * Stx + tcx * scy * Sw) * inv;
        float c  = (Syx - tcy * Ssx - scx * Sty + tcy * scx * Sw) * inv;
        float d  = (Syy - tcy * Ssy - scy * Sty + tcy * scy * Sw) * inv;

        float det = a * d - bb * c;
        float R00, R01, R10, R11, R22;
        if (det >= 0.0f) {           // closest rotation, R22 = +1
            float p = a + d, q = c - bb;
            float ir = 1.0f / sqrtf(fmaxf(p * p + q * q, 1e-30f));
            R00 = p * ir; R10 = q * ir; R01 = -R10; R11 = R00; R22 = 1.0f;
        } else {                     // closest reflection in-plane, R22 = -1
            float m = a - d, n = bb + c;
            float ir = 1.0f / sqrtf(fmaxf(m * m + n * n, 1e-30f));
            R00 = m * ir; R10 = n * ir; R01 = R10; R11 = -R00; R22 = -1.0f;
        }

        // u = src_centroid - R^T * tgt_centroid ; t = -R * u (z components = 0)
        float u0 = scx - (R00 * tcx + R10 * tcy);
        float u1 = scy - (R01 * tcx + R11 * tcy);
        float t0 = -(R00 * u0 + R01 * u1);
        float t1 = -(R10 * u0 + R11 * u1);

        float* oR = out + (size_t)b * 9;          // R transposed (0,2,1)
        oR[0] = R00; oR[1] = R10; oR[2] = 0.0f;
        oR[3] = R01; oR[4] = R11; oR[5] = 0.0f;
        oR[6] = 0.0f; oR[7] = 0.0f; oR[8] = R22;
        float* oT = out + (size_t)B * 9 + (size_t)b * 3;
        oT[0] = t0; oT[1] = t1; oT[2] = 0.0f;
    }
}

extern "C" void kernel_launch(void* const* d_in, const int* in_sizes, int n_in,
                              void* d_out, int out_size, void* d_ws, size_t ws_size,
                              hipStream_t stream) {
    const float* src  = (const float*)d_in[0];
    const float* tgt  = (const float*)d_in[1];
    const float* wts  = (const float*)d_in[2];
    const int*   flag = (const int*)d_in[3];
    float* out = (float*)d_out;
    const int B = out_size / 12;                 // 9 (R^T) + 3 (t) per batch
    const int N = in_sizes[2] / B;               // weights are [B,1,N]
    kabsch2d_kernel<<<dim3(B), dim3(THREADS), 0, stream>>>(src, tgt, wts, flag,
                                                           out, B, N);
}